// GATrEncoder_75514114998840
// MI455X (gfx1250) — compile-verified
//
#include <hip/hip_runtime.h>
#include <math.h>

// ---------------------------------------------------------------------------
// GATr encoder forward for gfx1250 (MI455X).  fp32 end-to-end; all channel-
// mixing GEMMs and attention GEMMs run on V_WMMA_F32_16X16X4_F32.
// LDS staging in the equi_linear kernels uses GLOBAL_LOAD_ASYNC_TO_LDS_B128
// (ASYNCcnt) -- the CDNA5 async global->LDS path.
//
// Input order assumption (jax pytree order: top-level dict insertion order,
// nested dicts sorted by key):
//  0 pts  1 view  2 tx
//  3 attn.k_w  4 attn.o_w  5 attn.q_w  6 attn.v_w
//  7 mlp.gp_l_w  8 mlp.gp_r_w  9 mlp.j_l_w  10 mlp.j_r_w
//  11 mlp.out_b  12 mlp.out_w
//  13 lin_in_b  14 lin_in_w  15 lin_out_b  16 lin_out_w
// ---------------------------------------------------------------------------

typedef __attribute__((ext_vector_type(2))) float v2f;
typedef __attribute__((ext_vector_type(8))) float v8f;

#define NB    4
#define NTOK  1024
#define TT    (NB*NTOK)   // 4096 tokens
#define ROW   1024        // 64 channels * 16 components
#define HROW  2048        // 128 channels * 16

// ---- PGA(3,0,1) blade tables (index order of the reference _BLADES list) ----
constexpr int   B_MASK[16] = {0,1,2,4,8,3,5,9,6,10,12,7,11,13,14,15};
constexpr int   M2I[16]    = {0,1,2,5,3,6,8,11,4,7,9,12,10,13,14,15};
constexpr int   B_GRADE[16]= {0,1,1,1,1,2,2,2,2,2,2,3,3,3,3,4};
constexpr int   B_PRE[16]  = {-1,0,-1,-1,-1,2,3,4,-1,-1,-1,8,9,10,-1,14};
constexpr float B_INV[16]  = {1,0,1,1,1,0,0,0,1,1,1,0,0,0,1,0};

__host__ __device__ constexpr int cpopc(int x){ int c=0; while(x){ c+=x&1; x>>=1; } return c; }
// sign of basis-blade product e_a * e_b (mask representation, swap count)
__host__ __device__ constexpr int psign(int a, int b){
  int s=0; for(int t=a>>1; t; t>>=1) s += cpopc(t&b);
  return (s&1) ? -1 : 1;
}
// sign of blade_i * complement(blade_i) (the D signed permutation)
__host__ __device__ constexpr float dsgn(int i){
  return (float)psign(B_MASK[i], 15 ^ B_MASK[i]);
}

// ---- fp32 WMMA wrapper ------------------------------------------------------
static __device__ inline v8f wmma4(v2f a, v2f b, v8f c){
  return __builtin_amdgcn_wmma_f32_16x16x4_f32(false, a, false, b, (short)0, c,
                                               false, false);
}

// ---- CDNA5 async global->LDS copy (16 bytes per lane), ASYNCcnt tracked ----
static __device__ inline void async_copy_b128(unsigned lds_byte_off,
                                              const float* gptr){
  asm volatile("global_load_async_to_lds_b128 %0, %1, off"
               :: "v"(lds_byte_off), "v"(gptr) : "memory");
}
static __device__ inline void wait_asynccnt0(){
  asm volatile("s_wait_asynccnt 0x0" ::: "memory");
}

// ---------------------------------------------------------------------------
// Weight relayout: w[Cout=64][Cin][9] -> Wt[r][Cout][Cin]  (A-operand friendly)
// ---------------------------------------------------------------------------
__global__ void k_wt(const float* __restrict__ w, float* __restrict__ wt, int cin){
  int i = blockIdx.x*blockDim.x + threadIdx.x;
  int tot = 9*64*cin;
  if(i >= tot) return;
  int c = i % cin; int m = (i/cin) % 64; int r = i/(cin*64);
  wt[i] = w[(m*cin + c)*9 + r];
}

// ---------------------------------------------------------------------------
// Embedding: pluecker ray + point -> x0[t][2][16]
// ---------------------------------------------------------------------------
__global__ void k_embed(const float* __restrict__ pts, const float* __restrict__ view,
                        const float* __restrict__ tx, float* __restrict__ x0){
  int t = blockIdx.x*blockDim.x + threadIdx.x;
  if(t >= TT) return;
  float px=pts[t*3+0], py=pts[t*3+1], pz=pts[t*3+2];
  float vx=view[t*3+0], vy=view[t*3+1], vz=view[t*3+2];
  float n = fmaxf(sqrtf(vx*vx+vy*vy+vz*vz), 1e-9f);
  vx/=n; vy/=n; vz/=n;
  float mx = py*vz - pz*vy, my = pz*vx - px*vz, mz = px*vy - py*vx;
  float r[16]; float q[16];
  #pragma unroll
  for(int j=0;j<16;++j){ r[j]=0.f; q[j]=0.f; }
  r[5]=mx; r[6]=my; r[7]=mz; r[8]=vz; r[9]=-vy; r[10]=vx;
  q[14]=1.f; q[13]=tx[t*3+0]; q[12]=-tx[t*3+1]; q[11]=tx[t*3+2];
  #pragma unroll
  for(int j=0;j<16;++j){ x0[t*32+j]=r[j]; x0[t*32+16+j]=q[j]; }
}

// ---------------------------------------------------------------------------
// lin_in: equi_linear with Cin=2 (tiny; pure VALU)
// ---------------------------------------------------------------------------
__global__ void k_linin(const float* __restrict__ x0, const float* __restrict__ w,
                        const float* __restrict__ b, float* __restrict__ X){
  int i = blockIdx.x*blockDim.x + threadIdx.x;
  if(i >= TT*64) return;
  int t = i >> 6, o = i & 63;
  const float* a0 = x0 + (size_t)t*32;
  const float* a1 = a0 + 16;
  const float* wo = w + o*18;   // [2][9]
  #pragma unroll
  for(int k=0;k<16;++k){
    int g = B_GRADE[k];
    float acc = wo[g]*a0[k] + wo[9+g]*a1[k];
    int p = B_PRE[k];
    if(p >= 0) acc += wo[4+g]*a0[p] + wo[9+4+g]*a1[p];
    if(k == 0) acc += b[o];
    X[(size_t)t*ROW + o*16 + k] = acc;
  }
}

// ---------------------------------------------------------------------------
// equi_layer_norm: one block per token, 64 threads (one per channel)
// ---------------------------------------------------------------------------
__global__ void k_ln(const float* __restrict__ X, float* __restrict__ Y){
  const int t = blockIdx.x, c = threadIdx.x;
  __shared__ float red[64];
  const float* xp = X + (size_t)t*ROW + c*16;
  float s = 0.f;
  #pragma unroll
  for(int j=0;j<16;++j){ float v = xp[j]; s += v*v*B_INV[j]; }
  red[c] = s; __syncthreads();
  for(int off=32; off>0; off>>=1){ if(c<off) red[c]+=red[c+off]; __syncthreads(); }
  const float sc = rsqrtf(red[0]*(1.0f/64.0f) + 1e-6f);
  float* yp = Y + (size_t)t*ROW + c*16;
  #pragma unroll
  for(int j=0;j<16;++j) yp[j] = xp[j]*sc;
}

// ---------------------------------------------------------------------------
// equi_linear via fp32 WMMA.  Block: 128 threads = 4 waves, one 16-token tile.
// Wave w produces out-channels [16w,16w+16).  X tile staged in LDS with the
// CDNA5 async global->LDS path (16B per lane, ASYNCcnt).
// Per component k:  Y_k = W_{grade(k)} X_k  (+ W_{4+grade(k)} X_{pre(k)} if e0∈k)
// ---------------------------------------------------------------------------
template<int CIN>
__global__ void __launch_bounds__(128, 1)
k_equilin(const float* __restrict__ Xin, int in_stride,
          const float* __restrict__ Wt, const float* __restrict__ bias,
          float* __restrict__ Yout, int out_stride, int choff, int accum)
{
  extern __shared__ float sX[];
  constexpr int PITCH = CIN*16 + 4;        // floats; row start stays 16B aligned
  const int t0  = blockIdx.x * 16;
  const int tid = threadIdx.x;

  // async-stage the 16-token tile: 16 x CIN x 16 floats, 16B chunks per lane
  {
    const int nchunk = 16*CIN*16/4;
    for(int i = tid; i < nchunk; i += blockDim.x){
      const int rem4 = i*4;
      const int tok  = rem4 / (CIN*16);
      const int rem  = rem4 - tok*(CIN*16);
      const unsigned ldsoff =
          (unsigned)(size_t)&sX[0] + (unsigned)((tok*PITCH + rem)*4);
      async_copy_b128(ldsoff, Xin + (size_t)(t0+tok)*in_stride + rem);
    }
  }
  wait_asynccnt0();
  __syncthreads();

  const int lane = tid & 31, wave = tid >> 5;
  const int half = lane >> 4, nl = lane & 15;
  const int m0 = wave * 16;

  #pragma unroll
  for(int k = 0; k < 16; ++k){
    v8f acc = {0,0,0,0,0,0,0,0};
    const int g = B_GRADE[k];
    {
      const float* wa = Wt + (size_t)(g*64 + m0 + nl)*CIN;
      const float* xb = sX + nl*PITCH + k;
      for(int c0 = 0; c0 < CIN; c0 += 4){
        v2f a, b;
        a.x = wa[c0 + 2*half];             // A: M=nl,  K = 2*half + {0,1}
        a.y = wa[c0 + 2*half + 1];
        b.x = xb[(c0 + 2*half    )*16];    // B: N=nl,  K = 2*half + {0,1}
        b.y = xb[(c0 + 2*half + 1)*16];
        acc = wmma4(a, b, acc);
      }
    }
    const int p = B_PRE[k];
    if(p >= 0){
      const float* wa = Wt + (size_t)((4+g)*64 + m0 + nl)*CIN;
      const float* xb = sX + nl*PITCH + p;
      for(int c0 = 0; c0 < CIN; c0 += 4){
        v2f a, b;
        a.x = wa[c0 + 2*half];
        a.y = wa[c0 + 2*half + 1];
        b.x = xb[(c0 + 2*half    )*16];
        b.y = xb[(c0 + 2*half + 1)*16];
        acc = wmma4(a, b, acc);
      }
    }
    // C/D layout: VGPR r, lanes0-15 -> M=r ; lanes16-31 -> M=8+r ; N=lane&15
    #pragma unroll
    for(int r = 0; r < 8; ++r){
      const int mloc = r + 8*half;
      float v = acc[r];
      if(bias != nullptr && k == 0) v += bias[m0 + mloc];
      float* dst = Yout + (size_t)(t0+nl)*out_stride + (choff + m0 + mloc)*16 + k;
      if(accum) *dst += v; else *dst = v;
    }
  }
}

// ---------------------------------------------------------------------------
// Flash attention, 1 wave per (batch, head, 16-query tile).  d = 128.
// INV mask + 1/8 scale folded into Q.  Online softmax with 16-lane butterflies.
// launch_bounds(32,1): keep qa/kb/oacc register-resident (no scratch spills);
// operand loads are grouped ahead of the WMMA chains so they clause.
// ---------------------------------------------------------------------------
__global__ void __launch_bounds__(32, 1)
k_attn(const float* __restrict__ Q, const float* __restrict__ K,
       const float* __restrict__ V, float* __restrict__ O)
{
  const int qt = blockIdx.x, h = blockIdx.y, b = blockIdx.z;
  const int lane = threadIdx.x;
  const int half = lane >> 4, nl = lane & 15;
  const int tq0 = b*NTOK + qt*16;
  const int doff = h*128;

  __shared__ float sP[16][17];

  // Q tile as 32 K-chunks (A operand), INV mask + softmax scale folded in
  v2f qa[32];
  {
    const float* qrow = Q + (size_t)(tq0+nl)*ROW + doff;
    #pragma unroll
    for(int kc = 0; kc < 32; ++kc){
      const int d0 = kc*4 + 2*half;
      qa[kc].x = qrow[d0  ] * (B_INV[ d0    & 15] * 0.125f);
      qa[kc].y = qrow[d0+1] * (B_INV[(d0+1) & 15] * 0.125f);
    }
  }

  float mrow[8], lrow[8];
  v8f oacc[8];
  #pragma unroll
  for(int r=0;r<8;++r){ mrow[r] = -3.0e38f; lrow[r] = 0.f;
                        oacc[r] = (v8f){0,0,0,0,0,0,0,0}; }

  for(int kt = 0; kt < 64; ++kt){
    const int tk0 = b*NTOK + kt*16;

    // ---- S = Q K^T : load all 32 B operands, then 32 chained WMMAs --------
    v2f kb[32];
    {
      const float* krow = K + (size_t)(tk0+nl)*ROW + doff;
      #pragma unroll
      for(int kc = 0; kc < 32; ++kc){
        const int d0 = kc*4 + 2*half;
        kb[kc].x = krow[d0  ];
        kb[kc].y = krow[d0+1];
      }
    }
    v8f s = {0,0,0,0,0,0,0,0};
    #pragma unroll
    for(int kc = 0; kc < 32; ++kc) s = wmma4(qa[kc], kb[kc], s);

    // ---- online softmax ---------------------------------------------------
    float alpha[8];
    #pragma unroll
    for(int r=0;r<8;++r){
      float v = s[r];
      float rm = v;
      #pragma unroll
      for(int o2=1;o2<16;o2<<=1) rm = fmaxf(rm, __shfl_xor(rm, o2, 32));
      const float mn = fmaxf(mrow[r], rm);
      alpha[r] = __expf(mrow[r] - mn);
      const float pv = __expf(v - mn);
      float rs = pv;
      #pragma unroll
      for(int o2=1;o2<16;o2<<=1) rs += __shfl_xor(rs, o2, 32);
      lrow[r] = lrow[r]*alpha[r] + rs;
      mrow[r] = mn;
      sP[r + 8*half][nl] = pv;              // D-layout -> LDS for relayout
    }
    __syncthreads();

    // P in A-layout (hoisted out of the dt loop)
    v2f pa[4];
    #pragma unroll
    for(int kc=0;kc<4;++kc){
      pa[kc].x = sP[nl][kc*4 + 2*half];
      pa[kc].y = sP[nl][kc*4 + 2*half + 1];
    }

    // ---- O = diag(alpha) O + P V ------------------------------------------
    #pragma unroll
    for(int dt=0; dt<8; ++dt){
      v2f vb[4];
      #pragma unroll
      for(int kc=0;kc<4;++kc){
        const int key = kc*4 + 2*half;
        vb[kc].x = V[(size_t)(tk0+key  )*ROW + doff + dt*16 + nl];
        vb[kc].y = V[(size_t)(tk0+key+1)*ROW + doff + dt*16 + nl];
      }
      #pragma unroll
      for(int r=0;r<8;++r) oacc[dt][r] *= alpha[r];
      #pragma unroll
      for(int kc=0;kc<4;++kc) oacc[dt] = wmma4(pa[kc], vb[kc], oacc[dt]);
    }
    __syncthreads();
  }

  #pragma unroll
  for(int dt=0; dt<8; ++dt){
    #pragma unroll
    for(int r=0;r<8;++r){
      O[(size_t)(tq0 + r + 8*half)*ROW + doff + dt*16 + nl] = oacc[dt][r] / lrow[r];
    }
  }
}

// ---------------------------------------------------------------------------
// ref[b] = mean over (N,C) of NORM[..., 15]
// ---------------------------------------------------------------------------
__global__ void k_ref(const float* __restrict__ X, float* __restrict__ ref){
  const int b = blockIdx.x, tid = threadIdx.x;
  __shared__ float red[256];
  float s = 0.f;
  for(int i=tid; i<NTOK*64; i+=256){
    const int n = i>>6, c = i&63;
    s += X[(size_t)(b*NTOK+n)*ROW + c*16 + 15];
  }
  red[tid]=s; __syncthreads();
  for(int off=128; off>0; off>>=1){ if(tid<off) red[tid]+=red[tid+off]; __syncthreads(); }
  if(tid==0) ref[b] = red[0] * (1.0f/(NTOK*64.0f));
}

// ---------------------------------------------------------------------------
// geometric product (ch<64) / ref * equi_join (ch>=64) + gelu gating -> H
// H second half already holds jr (read in place, then overwritten).
// Tables fully constant-folded by unrolling.
// ---------------------------------------------------------------------------
__global__ void k_gpjoin(const float* __restrict__ GL, const float* __restrict__ GR,
                         const float* __restrict__ JL, float* __restrict__ H,
                         const float* __restrict__ ref)
{
  const int i = blockIdx.x*blockDim.x + threadIdx.x;
  if(i >= TT*128) return;
  const int t = i >> 7, ch = i & 127;
  float out[16];
  #pragma unroll
  for(int k=0;k<16;++k) out[k]=0.f;

  if(ch < 64){
    const float* lp = GL + (size_t)t*ROW + ch*16;
    const float* rp = GR + (size_t)t*ROW + ch*16;
    float lv[16], rv[16];
    #pragma unroll
    for(int k=0;k<16;++k){ lv[k]=lp[k]; rv[k]=rp[k]; }
    #pragma unroll
    for(int ii=0;ii<16;++ii){
      #pragma unroll
      for(int jj=0;jj<16;++jj){
        const int mi = B_MASK[ii], mj = B_MASK[jj];
        if((mi & mj & 1) == 0){                      // e0*e0 = 0
          out[M2I[mi^mj]] += (float)psign(mi,mj) * lv[ii]*rv[jj];
        }
      }
    }
  } else {
    const float* lp = JL + (size_t)t*ROW + (ch-64)*16;
    const float* rp = H  + (size_t)t*HROW + ch*16;   // jr
    float lv[16], rv[16];
    #pragma unroll
    for(int k=0;k<16;++k){ lv[k]=lp[k]; rv[k]=rp[k]; }
    const float rr = ref[t >> 10];
    #pragma unroll
    for(int ii=0;ii<16;++ii){
      #pragma unroll
      for(int jj=0;jj<16;++jj){
        const int a  = 15 ^ B_MASK[ii];
        const int bm = 15 ^ B_MASK[jj];
        if((a & bm) == 0){                           // duals must be disjoint
          const int k = M2I[B_MASK[ii] & B_MASK[jj]];
          const float coef = dsgn(k)*dsgn(ii)*dsgn(jj)*(float)psign(a,bm);
          out[k] += coef * lv[ii]*rv[jj];
        }
      }
    }
    #pragma unroll
    for(int k=0;k<16;++k) out[k] *= rr;
  }
  const float g0 = out[0];
  const float gate = 0.5f*g0*(1.f + tanhf(0.7978845608028654f*
                                          (g0 + 0.044715f*g0*g0*g0)));
  float* dst = H + (size_t)t*HROW + ch*16;
  #pragma unroll
  for(int k=0;k<16;++k) dst[k] = gate*out[k];
}

// ---------------------------------------------------------------------------
// extract component 0 -> output (B,N,C)
// ---------------------------------------------------------------------------
__global__ void k_out(const float* __restrict__ Y, float* __restrict__ out){
  const int i = blockIdx.x*blockDim.x + threadIdx.x;
  if(i < TT*64) out[i] = Y[(size_t)(i>>6)*ROW + (i&63)*16];
}

// ---------------------------------------------------------------------------
extern "C" void kernel_launch(void* const* d_in, const int* in_sizes, int n_in,
                              void* d_out, int out_size, void* d_ws, size_t ws_size,
                              hipStream_t stream)
{
  (void)in_sizes; (void)n_in; (void)out_size; (void)ws_size;

  const float* pts   = (const float*)d_in[0];
  const float* view  = (const float*)d_in[1];
  const float* tx    = (const float*)d_in[2];
  const float* kw    = (const float*)d_in[3];
  const float* ow    = (const float*)d_in[4];
  const float* qw    = (const float*)d_in[5];
  const float* vw    = (const float*)d_in[6];
  const float* glw   = (const float*)d_in[7];
  const float* grw   = (const float*)d_in[8];
  const float* jlw   = (const float*)d_in[9];
  const float* jrw   = (const float*)d_in[10];
  const float* outb  = (const float*)d_in[11];
  const float* outw  = (const float*)d_in[12];
  const float* linib = (const float*)d_in[13];
  const float* liniw = (const float*)d_in[14];
  const float* linob = (const float*)d_in[15];
  const float* linow = (const float*)d_in[16];

  float* ws = (float*)d_ws;
  const size_t FT = (size_t)TT*ROW;          // 4,194,304 floats = 16 MB
  float* X   = ws;                           // residual
  float* NRM = ws + FT;                      // layernorm output
  float* Qb  = ws + 2*FT;
  float* Kb  = ws + 3*FT;
  float* Vb  = ws + 4*FT;
  float* Hb  = ws + 5*FT;                    // 2*FT (also attention output)
  float* X0  = ws + 7*FT;                    // embeddings (T,2,16)
  float* REF = X0 + (size_t)TT*32;
  float* WQ  = REF + 16;
  const size_t WSZ = 9*64*64;
  float* WK  = WQ  + WSZ; float* WV  = WK  + WSZ; float* WO  = WV  + WSZ;
  float* WGL = WO  + WSZ; float* WGR = WGL + WSZ; float* WJL = WGR + WSZ;
  float* WJR = WJL + WSZ; float* WLO = WJR + WSZ;
  float* WOUT= WLO + WSZ;                    // 9*64*128

  auto wt = [&](const float* src, float* dst, int cin){
    int tot = 9*64*cin;
    k_wt<<<(tot+255)/256, 256, 0, stream>>>(src, dst, cin);
  };
  wt(qw, WQ, 64);  wt(kw, WK, 64);  wt(vw, WV, 64);  wt(ow, WO, 64);
  wt(glw, WGL, 64); wt(grw, WGR, 64); wt(jlw, WJL, 64); wt(jrw, WJR, 64);
  wt(linow, WLO, 64); wt(outw, WOUT, 128);

  k_embed<<<(TT+255)/256, 256, 0, stream>>>(pts, view, tx, X0);
  k_linin<<<(TT*64+255)/256, 256, 0, stream>>>(X0, liniw, linib, X);

  const size_t LDS64  = (size_t)16*( 64*16+4)*sizeof(float);
  const size_t LDS128 = (size_t)16*(128*16+4)*sizeof(float);

  // ---- attention sub-block -------------------------------------------------
  k_ln<<<TT, 64, 0, stream>>>(X, NRM);
  k_equilin<64><<<TT/16, 128, LDS64, stream>>>(NRM, ROW, WQ, nullptr, Qb, ROW, 0, 0);
  k_equilin<64><<<TT/16, 128, LDS64, stream>>>(NRM, ROW, WK, nullptr, Kb, ROW, 0, 0);
  k_equilin<64><<<TT/16, 128, LDS64, stream>>>(NRM, ROW, WV, nullptr, Vb, ROW, 0, 0);
  k_attn<<<dim3(64,8,4), 32, 0, stream>>>(Qb, Kb, Vb, Hb);
  k_equilin<64><<<TT/16, 128, LDS64, stream>>>(Hb, ROW, WO, nullptr, X, ROW, 0, 1);

  // ---- geometric MLP sub-block --------------------------------------------
  k_ln<<<TT, 64, 0, stream>>>(X, NRM);
  k_ref<<<NB, 256, 0, stream>>>(NRM, REF);
  k_equilin<64><<<TT/16, 128, LDS64, stream>>>(NRM, ROW, WGL, nullptr, Qb, ROW, 0, 0);
  k_equilin<64><<<TT/16, 128, LDS64, stream>>>(NRM, ROW, WGR, nullptr, Kb, ROW, 0, 0);
  k_equilin<64><<<TT/16, 128, LDS64, stream>>>(NRM, ROW, WJL, nullptr, Vb, ROW, 0, 0);
  k_equilin<64><<<TT/16, 128, LDS64, stream>>>(NRM, ROW, WJR, nullptr, Hb, HROW, 64, 0);
  k_gpjoin<<<(TT*128+255)/256, 256, 0, stream>>>(Qb, Kb, Vb, Hb, REF);
  k_equilin<128><<<TT/16, 128, LDS128, stream>>>(Hb, HROW, WOUT, outb, X, ROW, 0, 1);

  // ---- output head ---------------------------------------------------------
  k_equilin<64><<<TT/16, 128, LDS64, stream>>>(X, ROW, WLO, linob, NRM, ROW, 0, 0);
  k_out<<<(TT*64+255)/256, 256, 0, stream>>>(NRM, (float*)d_out);
}